// SAGEConv_47974784697088
// MI455X (gfx1250) — compile-verified
//
#include <hip/hip_runtime.h>

// SAGEConv (mean aggregator, bias=False) for MI455X / gfx1250 (wave32, WMMA).
//   out = h @ W_self + (segment_mean over in-edges of h[src]) @ W_neigh
// N=100000 nodes, E=1600000 edges, D_IN=D_OUT=128.
//
// Phase 1: zero workspace (neigh_sum[N][128] + deg[N]).
// Phase 2: one wave32 per edge; 4 coalesced fp32 atomicAdds per lane into
//          neigh_sum[dst] (h fits in 192MB L2, so the gather is L2-resident),
//          lane 0 accumulates degree.
// Phase 3: fused GEMM  [h | neigh_sum/deg] (K=256) x [W_self ; W_neigh]
//          using V_WMMA_F32_16X16X4_F32 (fp32-exact path, matches reference
//          precision). 64 rows/block, 32 waves, each wave one 16x16 tile.

typedef __attribute__((ext_vector_type(2))) float v2f;
typedef __attribute__((ext_vector_type(8))) float v8f;

#define N_NODES_C 100000
#define N_EDGES_C 1600000
#define D_C 128              // D_IN == D_OUT
#define ROWS_PB 64           // node rows per block in GEMM
#define A_PAD 2              // LDS row padding (floats) -> conflict-free ds_load_b64

__global__ void zero_kernel(float* __restrict__ p, long long n) {
  long long i = (long long)blockIdx.x * blockDim.x + threadIdx.x;
  if (i < n) p[i] = 0.0f;
}

// One wave (32 lanes) per edge: scatter h[src] into neigh_sum[dst], count deg.
__global__ __launch_bounds__(256)
void scatter_kernel(const float* __restrict__ h,
                    const int* __restrict__ src,
                    const int* __restrict__ dst,
                    float* __restrict__ nsum,
                    float* __restrict__ deg) {
  long long t = (long long)blockIdx.x * blockDim.x + threadIdx.x;
  int e    = (int)(t >> 5);
  int lane = (int)(t & 31);
  if (e >= N_EDGES_C) return;
  int s = src[e];
  int d = dst[e];
  const float* hs = h    + (long long)s * D_C;
  float*       ns = nsum + (long long)d * D_C;
#pragma unroll
  for (int j = 0; j < D_C / 32; ++j) {
    atomicAdd(ns + lane + 32 * j, hs[lane + 32 * j]);   // wave-coalesced 128B
  }
  if (lane == 0) atomicAdd(deg + d, 1.0f);
}

// Fused finalize + GEMM. Block: 1024 threads = 32 waves. Wave w computes the
// 16x16 output tile (rowTile = w>>3, colTile = w&7) via 64 accumulating
// v_wmma_f32_16x16x4_f32 ops over K=256 ( [h | h_neigh] x [Wself ; Wneigh] ).
__global__ __launch_bounds__(1024)
void sage_gemm_kernel(const float* __restrict__ h,
                      const float* __restrict__ nsum,
                      const float* __restrict__ deg,
                      const float* __restrict__ Wself,
                      const float* __restrict__ Wneigh,
                      float* __restrict__ out) {
  __shared__ float A[ROWS_PB][2 * D_C + A_PAD];   // 64 x 258 floats = 66 KB LDS

  const int rowBase = blockIdx.x * ROWS_PB;
  const int tid = threadIdx.x;

  // Stage A = [ h_row(0:128) | neigh_mean_row(0:128) ] for 64 rows.
  for (int idx = tid; idx < ROWS_PB * 2 * D_C; idx += 1024) {
    int r = idx >> 8;            // / 256
    int k = idx & 255;           // % 256
    int node = rowBase + r;
    float val = 0.0f;
    if (node < N_NODES_C) {
      if (k < D_C) {
        val = h[(long long)node * D_C + k];
      } else {
        float rd = 1.0f / fmaxf(deg[node], 1.0f);
        val = nsum[(long long)node * D_C + (k - D_C)] * rd;
      }
    }
    A[r][k] = val;
  }
  __syncthreads();

  const int wave    = tid >> 5;       // 0..31
  const int lane    = tid & 31;
  const int rowTile = wave >> 3;      // 0..3  (16 rows each)
  const int colTile = wave & 7;       // 0..7  (16 cols each)
  const int m       = lane & 15;      // row within A-frag / col within B-frag
  const int hi      = lane >> 4;      // K sub-pair select (ISA A/B layout)
  const int col     = colTile * 16 + m;

  const float* Arow = &A[rowTile * 16 + m][0];
  v8f acc = {};

#pragma unroll
  for (int half = 0; half < 2; ++half) {
    const float* W  = (half == 0) ? Wself : Wneigh;   // row-major [128][128]
    const float* Ah = Arow + half * D_C;
#pragma unroll 8
    for (int k0 = 0; k0 < D_C; k0 += 4) {
      int ka = k0 + 2 * hi;
      // A-frag 16x4: lanes0-15 K={k0,k0+1}, lanes16-31 K={k0+2,k0+3}
      v2f a = *(const v2f*)(Ah + ka);                  // ds_load_b64, no bank conflict
      // B-frag 4x16: V0 row ka, V1 row ka+1, striped across lanes by column
      v2f b;
      b.x = W[(long long)ka * D_C + col];
      b.y = W[(long long)(ka + 1) * D_C + col];
      acc = __builtin_amdgcn_wmma_f32_16x16x4_f32(
          /*neg_a=*/false, a, /*neg_b=*/false, b,
          /*c_mod=*/(short)0, acc, /*reuse_a=*/false, /*reuse_b=*/false);
    }
  }

  // C/D layout: VGPR r -> row (8*hi + r) of the 16x16 tile.
  const int row0 = rowBase + rowTile * 16 + hi * 8;
#pragma unroll
  for (int r = 0; r < 8; ++r) {
    int row = row0 + r;
    if (row < N_NODES_C)
      out[(long long)row * D_C + col] = acc[r];
  }
}

extern "C" void kernel_launch(void* const* d_in, const int* in_sizes, int n_in,
                              void* d_out, int out_size, void* d_ws, size_t ws_size,
                              hipStream_t stream) {
  const float* h      = (const float*)d_in[0];
  const int*   src    = (const int*)d_in[1];   // int32 per harness integer convention
  const int*   dst    = (const int*)d_in[2];
  const float* Wself  = (const float*)d_in[3];
  const float* Wneigh = (const float*)d_in[4];
  float*       out    = (float*)d_out;

  float* nsum = (float*)d_ws;                           // N*128 floats
  float* deg  = nsum + (long long)N_NODES_C * D_C;      // N floats

  // Phase 1: zero accumulators.
  long long nz = (long long)N_NODES_C * D_C + N_NODES_C;
  zero_kernel<<<(unsigned)((nz + 255) / 256), 256, 0, stream>>>(nsum, nz);

  // Phase 2: edge scatter (one wave32 per edge).
  long long tscat = (long long)N_EDGES_C * 32;
  scatter_kernel<<<(unsigned)((tscat + 255) / 256), 256, 0, stream>>>(
      h, src, dst, nsum, deg);

  // Phase 3: fused normalize + dual GEMM via fp32 WMMA.
  int nblk = (N_NODES_C + ROWS_PB - 1) / ROWS_PB;
  sage_gemm_kernel<<<nblk, 1024, 0, stream>>>(h, nsum, deg, Wself, Wneigh, out);
}